// NodeGNN_56435870269829
// MI455X (gfx1250) — compile-verified
//
#include <hip/hip_runtime.h>

typedef float v2f __attribute__((ext_vector_type(2)));
typedef float v8f __attribute__((ext_vector_type(8)));

#define KDIM 128  // inner dim of both SAGE layers

// ---------------------------------------------------------------------------
// Zero a float range
__global__ void zero_kernel(float* __restrict__ p, long long n) {
    long long i = (long long)blockIdx.x * blockDim.x + threadIdx.x;
    if (i < n) p[i] = 0.0f;
}

// ---------------------------------------------------------------------------
// deg[dst[e]] += 1
__global__ void deg_kernel(const int* __restrict__ dst, float* deg, int nedges) {
    int e = blockIdx.x * blockDim.x + threadIdx.x;
    if (e < nedges) atomicAdd(&deg[dst[e]], 1.0f);
}

// ---------------------------------------------------------------------------
// agg[dst[e]][:] += feat[src[e]][:]   (one wave covers F floats of one edge
// for F=128; coalesced float4 gather, contiguous atomics per dst row)
template <int F>
__global__ void scatter_kernel(const float* __restrict__ feat,
                               const int* __restrict__ src,
                               const int* __restrict__ dst,
                               float* agg, int nedges) {
    const int CH = F / 4;  // float4 chunks per edge
    long long gid = (long long)blockIdx.x * blockDim.x + threadIdx.x;
    if (gid >= (long long)nedges * CH) return;
    int e = (int)(gid / CH);
    int c = (int)(gid % CH);
    int s = src[e], d = dst[e];
    float4 v = *(const float4*)(feat + (size_t)s * F + c * 4);
    float* p = agg + (size_t)d * F + c * 4;
    atomicAdd(p + 0, v.x);
    atomicAdd(p + 1, v.y);
    atomicAdd(p + 2, v.z);
    atomicAdd(p + 3, v.w);
}

// ---------------------------------------------------------------------------
// Y[m][n] = sum_k X[m][k] * W[n][k]            (SAGE == false)
// Y[m][n] = relu(agg[m][n]/max(deg[m],1) + bias[n] + sum_k X[m][k]*W[n][k])
//                                              (SAGE == true; Y may alias agg)
// Block = 128 threads = 4 waves; wave w handles rows [blk*64 + w*16, +16).
// f32 WMMA 16x16x4; A fragments held in registers across the whole kernel,
// W staged in LDS in 64-row chunks (stride 132 floats: 16B-aligned float4
// stores, bank-conflict-free b64 fragment reads).
template <int NOUT, bool SAGE>
__global__ __launch_bounds__(128) void gemm_wmma_kernel(
    const float* __restrict__ X,     // [nrows, 128]
    const float* __restrict__ W,     // [NOUT, 128] row-major
    const float* bias,               // [NOUT]        (SAGE)
    const float* agg,                // [nrows, NOUT] (SAGE)
    const float* __restrict__ deg,   // [nrows]       (SAGE)
    float* Y,                        // [nrows, NOUT]
    int nrows) {
    __shared__ float sW[64 * 132];  // 33 KB

    const int lane = threadIdx.x & 31;
    const int wave = threadIdx.x >> 5;
    const int lo = lane & 15;
    const int hi = lane >> 4;

    const int m0 = blockIdx.x * 64 + wave * 16;

    // --- preload A fragments: lane holds X[row][4*kk + 2*hi .. +1] ---
    int arow = m0 + lo;
    if (arow >= nrows) arow = nrows - 1;  // clamp (stores are predicated)
    const float* aptr = X + (size_t)arow * KDIM + 2 * hi;
    v2f afrag[KDIM / 4];
#pragma unroll
    for (int kk = 0; kk < KDIM / 4; ++kk)
        afrag[kk] = *(const v2f*)(aptr + 4 * kk);

    const int bbase = lo * 132 + 2 * hi;  // per-lane B-fragment base in LDS

    for (int chunk = 0; chunk < NOUT / 64; ++chunk) {
        __syncthreads();  // WAR guard on sW for chunk > 0
        // cooperative load of W rows [chunk*64, chunk*64+64) into LDS
        const float* wsrc = W + (size_t)chunk * 64 * KDIM;
        for (int idx = threadIdx.x; idx < 64 * (KDIM / 4); idx += 128) {
            int r = idx >> 5;           // KDIM/4 == 32 float4 per row
            int c4 = (idx & 31) << 2;
            *(float4*)(&sW[r * 132 + c4]) = *(const float4*)(wsrc + r * KDIM + c4);
        }
        __syncthreads();

#pragma unroll
        for (int t = 0; t < 4; ++t) {  // 4 N-tiles of 16 per chunk
            v8f acc = {};
            const float* bp = &sW[t * 16 * 132 + bbase];
#pragma unroll
            for (int kk = 0; kk < KDIM / 4; ++kk) {
                v2f bfrag = *(const v2f*)(bp + 4 * kk);
                acc = __builtin_amdgcn_wmma_f32_16x16x4_f32(
                    false, afrag[kk], false, bfrag, (short)0, acc, false, false);
            }
            // C/D layout: VGPR j -> (M = j + 8*hi, N = lo)
            int ncol = chunk * 64 + t * 16 + lo;
#pragma unroll
            for (int j = 0; j < 8; ++j) {
                int m = m0 + j + 8 * hi;
                if (m < nrows) {
                    float v = acc[j];
                    if (SAGE) {
                        float d = deg[m];
                        d = d > 1.0f ? d : 1.0f;
                        v += agg[(size_t)m * NOUT + ncol] / d + bias[ncol];
                        v = v > 0.0f ? v : 0.0f;
                    }
                    Y[(size_t)m * NOUT + ncol] = v;
                }
            }
        }
    }
}

// ---------------------------------------------------------------------------
// out[n][o] = bfc[o] + sum_j h2[n][j] * Wfc[o][j],  j<64, o<2.
// One wave per node; lane handles 2 features; shuffle reduction.
__global__ __launch_bounds__(128) void fc_kernel(const float* __restrict__ h2,
                                                 const float* __restrict__ Wfc,
                                                 const float* __restrict__ bfc,
                                                 float* __restrict__ out,
                                                 int nrows) {
    int lane = threadIdx.x & 31;
    int wave = threadIdx.x >> 5;
    int n = blockIdx.x * 4 + wave;
    if (n >= nrows) return;
    float2 hv = *(const float2*)(h2 + (size_t)n * 64 + lane * 2);
    float2 w0 = *(const float2*)(Wfc + lane * 2);
    float2 w1 = *(const float2*)(Wfc + 64 + lane * 2);
    float p0 = hv.x * w0.x + hv.y * w0.y;
    float p1 = hv.x * w1.x + hv.y * w1.y;
#pragma unroll
    for (int off = 16; off > 0; off >>= 1) {
        p0 += __shfl_down(p0, off, 32);
        p1 += __shfl_down(p1, off, 32);
    }
    if (lane == 0) {
        out[(size_t)n * 2 + 0] = p0 + bfc[0];
        out[(size_t)n * 2 + 1] = p1 + bfc[1];
    }
}

// ---------------------------------------------------------------------------
extern "C" void kernel_launch(void* const* d_in, const int* in_sizes, int n_in,
                              void* d_out, int out_size, void* d_ws, size_t ws_size,
                              hipStream_t stream) {
    (void)n_in; (void)out_size; (void)ws_size;
    const float* x   = (const float*)d_in[0];
    const int*   ei  = (const int*)d_in[1];
    const float* W1l = (const float*)d_in[2];
    const float* b1  = (const float*)d_in[3];
    const float* W1r = (const float*)d_in[4];
    const float* W2l = (const float*)d_in[5];
    const float* b2  = (const float*)d_in[6];
    const float* W2r = (const float*)d_in[7];
    const float* Wfc = (const float*)d_in[8];
    const float* bfc = (const float*)d_in[9];
    float* out = (float*)d_out;

    const int N = in_sizes[0] / 128;
    const int E = in_sizes[1] / 2;
    const int* src = ei;
    const int* dst = ei + E;

    // workspace layout (floats): buf0[N*128] | buf2[N*128] | deg[N]
    float* buf0 = (float*)d_ws;                  // xl; later hl(0:N*64) + agg2(N*64:N*128)
    float* buf2 = buf0 + (size_t)N * 128;        // agg1 -> h (in place)
    float* degp = buf0 + (size_t)N * 256;

    // 1. zero agg1 + deg (contiguous)
    {
        long long n = (long long)N * 128 + N;
        zero_kernel<<<(int)((n + 255) / 256), 256, 0, stream>>>(buf2, n);
    }
    // 2. degrees
    deg_kernel<<<(E + 255) / 256, 256, 0, stream>>>(dst, degp, E);

    const int gblk = (N + 63) / 64;
    // 3. xl = x @ W1l^T
    gemm_wmma_kernel<128, false><<<gblk, 128, 0, stream>>>(
        x, W1l, nullptr, nullptr, nullptr, buf0, N);
    // 4. agg1 += xl[src] per edge
    {
        long long tot = (long long)E * 32;
        scatter_kernel<128><<<(int)((tot + 255) / 256), 256, 0, stream>>>(
            buf0, src, dst, buf2, E);
    }
    // 5. h = relu(agg1/deg + b1 + x @ W1r^T)  (in place over agg1)
    gemm_wmma_kernel<128, true><<<gblk, 128, 0, stream>>>(
        x, W1r, b1, buf2, degp, buf2, N);
    // 6. hl = h @ W2l^T -> buf0[0:N*64]
    gemm_wmma_kernel<64, false><<<gblk, 128, 0, stream>>>(
        buf2, W2l, nullptr, nullptr, nullptr, buf0, N);
    // 7. zero agg2 = buf0[N*64:N*128]
    {
        long long n = (long long)N * 64;
        zero_kernel<<<(int)((n + 255) / 256), 256, 0, stream>>>(buf0 + (size_t)N * 64, n);
    }
    // 8. agg2 += hl[src] per edge (64-wide: half the scatter traffic)
    {
        long long tot = (long long)E * 16;
        scatter_kernel<64><<<(int)((tot + 255) / 256), 256, 0, stream>>>(
            buf0, src, dst, buf0 + (size_t)N * 64, E);
    }
    // 9. h2 = relu(agg2/deg + b2 + h @ W2r^T) -> buf0[0:N*64] (hl dead)
    gemm_wmma_kernel<64, true><<<gblk, 128, 0, stream>>>(
        buf2, W2r, b2, buf0 + (size_t)N * 64, degp, buf0, N);
    // 10. out = h2 @ Wfc^T + bfc
    fc_kernel<<<(N + 3) / 4, 128, 0, stream>>>(buf0, Wfc, bfc, out, N);
}